// PerFilterTrace_17059610100015
// MI455X (gfx1250) — compile-verified
//
#include <hip/hip_runtime.h>

// EWMA scan z_t = LAM*z_{t-1} + (1-LAM)*x_t over [B=16, L=8192, K=256] f32.
// 3-pass chunked linear recurrence; pass 3 does 16x16 time-channel tiles via
// V_WMMA_F32_16X16X4_F32 (4 chained K=4 slabs of the lower-triangular lambda
// matrix + 1 carry WMMA).

#define LAM    0.95f
#define ONEM   0.05f      // 1 - LAM
#define BSZ    16
#define LSEQ   8192
#define KSZ    256
#define NCH    64         // chunks along L
#define LC     (LSEQ / NCH)   // 128 steps per chunk
#define NKG    (KSZ / 16)     // 16 channel groups of 16

typedef float v2f __attribute__((ext_vector_type(2)));
typedef float v8f __attribute__((ext_vector_type(8)));

// ---------------- Pass 1: per-chunk local-scan carries ----------------
// thread (b, c, k): s = local scan (zero init) over LC steps; store final.
__global__ void ewma_pass1_carries(const float* __restrict__ x,
                                   float* __restrict__ carry) {
    const int k = threadIdx.x;            // 0..255
    const int c = blockIdx.x % NCH;
    const int b = blockIdx.x / NCH;
    const float* xp = x + ((size_t)b * LSEQ + (size_t)c * LC) * KSZ + k;
    float z = 0.0f;
#pragma unroll 8
    for (int i = 0; i < LC; ++i) {
        z = LAM * z + ONEM * xp[(size_t)i * KSZ];
    }
    carry[((size_t)b * NCH + c) * KSZ + k] = z;
}

// ---------------- Pass 2: combine chunk carries -> chunk carry-ins ----------------
// thread (b, k): Z_in(c) = lam^LC * Z_in(c-1) + s_{c-1}, Z_in(0) = 0.
__global__ void ewma_pass2_scan(const float* __restrict__ carry,
                                float* __restrict__ zin) {
    const int k = threadIdx.x;            // 0..255
    const int b = blockIdx.x;             // 0..15
    const float pw = __powf(LAM, (float)LC);
    float acc = 0.0f;
    for (int c = 0; c < NCH; ++c) {
        const size_t idx = ((size_t)b * NCH + c) * KSZ + k;
        zin[idx] = acc;
        acc = pw * acc + carry[idx];
    }
}

// ---------------- Pass 3: WMMA tile scan, write full trajectory ----------------
// One wave per (b, kg, c). Tile = 16 time steps x 16 channels:
//   D[i][n] = sum_{j<=i} (1-lam)*lam^{i-j} * X[j][n]  +  lam^{i+1} * z_in[n]
// computed as 5 chained V_WMMA_F32_16X16X4_F32 accumulations.
__global__ void __launch_bounds__(128)
ewma_pass3_wmma(const float* __restrict__ x,
                const float* __restrict__ zin,
                float* __restrict__ out) {
    const int wave = blockIdx.x * (blockDim.x >> 5) + (threadIdx.x >> 5);
    const int lane = threadIdx.x & 31;
    const int c  = wave % NCH;
    const int kg = (wave / NCH) % NKG;
    const int b  = wave / (NCH * NKG);
    const int k0 = kg * 16;

    const int  m  = lane & 15;            // row index (A/M), col index (B/N)
    const bool lo = lane < 16;

    // A-matrix slabs: ISA 16x4 f32 layout -> lanes<16 hold cols {4q+0,4q+1},
    // lanes>=16 hold cols {4q+2,4q+3} in (v0,v1). M[i][j]=(1-lam)*lam^(i-j), j<=i.
    const int jj0 = lo ? 0 : 2;
    v2f Aq[4];
#pragma unroll
    for (int q = 0; q < 4; ++q) {
        const int c0 = 4 * q + jj0;
        const int c1 = c0 + 1;
        float a0 = (m >= c0) ? ONEM * __powf(LAM, (float)(m - c0)) : 0.0f;
        float a1 = (m >= c1) ? ONEM * __powf(LAM, (float)(m - c1)) : 0.0f;
        Aq[q] = (v2f){a0, a1};
    }
    // Carry column vector lam^{m+1} living in K-column 0 only.
    const v2f Acar = (v2f){ lo ? __powf(LAM, (float)(m + 1)) : 0.0f, 0.0f };

    // Initial carry-in z_{chunkstart-1} for channels k0..k0+15.
    float zval = zin[((size_t)b * NCH + c) * KSZ + k0 + m];

    const size_t base = ((size_t)b * LSEQ + (size_t)c * LC) * KSZ + k0;
    const float* xp = x + base;
    float* op = out + base;

    const int r0   = lo ? 0 : 2;   // B-matrix: row-in-slab held by this half-wave
    const int rofD = lo ? 0 : 8;   // D-matrix: M offset for this half-wave

    for (int tile = 0; tile < LC / 16; ++tile) {
        const int t0 = tile * 16;

        // B for carry WMMA: z_in in row K=0 (lanes<16), rows K=1..3 zero.
        const v2f Bcar = (v2f){ lo ? zval : 0.0f, 0.0f };
        v8f acc = {};
        acc = __builtin_amdgcn_wmma_f32_16x16x4_f32(
            false, Acar, false, Bcar, (short)0, acc, false, false);

#pragma unroll
        for (int q = 0; q < 4; ++q) {
            const int row = t0 + 4 * q + r0;
            v2f Bx;
            Bx.x = xp[(size_t)row * KSZ + m];
            Bx.y = xp[(size_t)(row + 1) * KSZ + m];
            acc = __builtin_amdgcn_wmma_f32_16x16x4_f32(
                false, Aq[q], false, Bx, (short)0, acc, false, false);
        }

        // Store D: VGPR r holds rows (r | r+8) striped across half-waves.
#pragma unroll
        for (int r = 0; r < 8; ++r) {
            op[(size_t)(t0 + r + rofD) * KSZ + m] = acc[r];
        }

        // Next tile's carry = D row 15 = acc[7] in lanes 16..31 (N = lane-16).
        zval = __shfl(acc[7], 16 + m, 32);
    }
}

extern "C" void kernel_launch(void* const* d_in, const int* in_sizes, int n_in,
                              void* d_out, int out_size, void* d_ws, size_t ws_size,
                              hipStream_t stream) {
    (void)in_sizes; (void)n_in; (void)out_size; (void)ws_size;
    const float* x  = (const float*)d_in[0];
    float* out      = (float*)d_out;

    // Workspace: [0, 1MB) chunk carries, [1MB, 2MB) chunk carry-ins.
    float* carry = (float*)d_ws;
    float* zin   = carry + (size_t)BSZ * NCH * KSZ;

    ewma_pass1_carries<<<dim3(BSZ * NCH), dim3(KSZ), 0, stream>>>(x, carry);
    ewma_pass2_scan  <<<dim3(BSZ),       dim3(KSZ), 0, stream>>>(carry, zin);

    const int waves  = BSZ * NKG * NCH;          // 16*16*64 = 16384 waves
    const int blockT = 128;                      // 4 waves / block
    ewma_pass3_wmma<<<dim3(waves / (blockT / 32)), dim3(blockT), 0, stream>>>(x, zin, out);
}